// Struct3Model_27625229648374
// MI455X (gfx1250) — compile-verified
//
#include <hip/hip_runtime.h>

typedef __attribute__((ext_vector_type(2))) float v2f;
typedef __attribute__((ext_vector_type(8))) float v8f;

#define QS 1204
#define FD 7
#define NCAT 600
#define CAT_STRIDE 601
#define H1 128
#define H1_STRIDE 129
#define H2 32
#define H2_STRIDE 33
#define ROWS_PER_BLOCK 16

__global__ __launch_bounds__(256) void fused_struct3_kernel(
    const float* __restrict__ x,       // (16384, 1204, 7)
    const float* __restrict__ w_step,  // (7,)
    const float* __restrict__ b_step,  // (1,)
    const float* __restrict__ wA,      // (600, 128)
    const float* __restrict__ bA,      // (128,)
    const float* __restrict__ wB,      // (128, 32)
    const float* __restrict__ bB,      // (32,)
    const float* __restrict__ wC,      // (32, 2)
    const float* __restrict__ bC,      // (2,)
    float* __restrict__ out)           // (16384, 2)
{
    __shared__ float s_out[QS];                        // per-row dot results
    __shared__ float s_cat[ROWS_PER_BLOCK * CAT_STRIDE];
    __shared__ float s_h1[ROWS_PER_BLOCK * H1_STRIDE];
    __shared__ float s_h2[ROWS_PER_BLOCK * H2_STRIDE];

    const int tid  = threadIdx.x;
    const int wave = tid >> 5;
    const int lane = tid & 31;
    const int half = lane >> 4;   // which half-wave (selects K pair / M half)
    const int lm   = lane & 15;   // M index for A-frag, N index for B-frag
    const int b0   = blockIdx.x * ROWS_PER_BLOCK;

    const float ws0 = w_step[0], ws1 = w_step[1], ws2 = w_step[2],
                ws3 = w_step[3], ws4 = w_step[4], ws5 = w_step[5],
                ws6 = w_step[6];
    const float bs = b_step[0];

    // ---------------- Stage 1: fc_step dot + windowed max pooling ----------
    for (int r = 0; r < ROWS_PER_BLOCK; ++r) {
        const float* xr = x + (size_t)(b0 + r) * (QS * FD);
        for (int q = tid; q < QS; q += 256) {
            const float* p = xr + q * FD;
            float acc = bs;
            acc += p[0] * ws0; acc += p[1] * ws1; acc += p[2] * ws2;
            acc += p[3] * ws3; acc += p[4] * ws4; acc += p[5] * ws5;
            acc += p[6] * ws6;
            s_out[q] = acc;
        }
        __syncthreads();
        for (int c = tid; c < NCAT; c += 256) {
            const int chunk = c / 5;
            const int off   = c - chunk * 5;
            const int st    = chunk * 10 + off;
            float m = s_out[st];
            #pragma unroll
            for (int w = 1; w < 10; ++w) m = fmaxf(m, s_out[st + w]);
            s_cat[r * CAT_STRIDE + c] = m;
        }
        __syncthreads();
    }

    // ---------------- GEMM1: (16x600) @ (600x128), bias + ReLU -------------
    // Each wave owns one 16-wide N tile. K in steps of 4 via f32 WMMA.
    {
        const int n0 = wave * 16;
        v8f c = {};
        const float* a_row = &s_cat[lm * CAT_STRIDE];
        const float* b_col = wA + n0 + lm;
        for (int k = 0; k < NCAT; k += 4) {
            const int ka = k + 2 * half;   // this half-wave supplies K=ka, ka+1
            v2f a, b;
            a.x = a_row[ka];
            a.y = a_row[ka + 1];
            b.x = b_col[(size_t)ka * H1];
            b.y = b_col[(size_t)(ka + 1) * H1];
            c = __builtin_amdgcn_wmma_f32_16x16x4_f32(
                    false, a, false, b, (short)0, c, false, false);
        }
        const float bias = bA[n0 + lm];
        #pragma unroll
        for (int v = 0; v < 8; ++v) {
            const int m = v + (half ? 8 : 0);
            s_h1[m * H1_STRIDE + n0 + lm] = fmaxf(c[v] + bias, 0.0f);
        }
    }
    __syncthreads();

    // ---------------- GEMM2: (16x128) @ (128x32), bias + ReLU --------------
    if (wave < 2) {
        const int n0 = wave * 16;
        v8f c = {};
        const float* a_row = &s_h1[lm * H1_STRIDE];
        const float* b_col = wB + n0 + lm;
        #pragma unroll 4
        for (int k = 0; k < H1; k += 4) {
            const int ka = k + 2 * half;
            v2f a, b;
            a.x = a_row[ka];
            a.y = a_row[ka + 1];
            b.x = b_col[ka * H2];
            b.y = b_col[(ka + 1) * H2];
            c = __builtin_amdgcn_wmma_f32_16x16x4_f32(
                    false, a, false, b, (short)0, c, false, false);
        }
        const float bias = bB[n0 + lm];
        #pragma unroll
        for (int v = 0; v < 8; ++v) {
            const int m = v + (half ? 8 : 0);
            s_h2[m * H2_STRIDE + n0 + lm] = fmaxf(c[v] + bias, 0.0f);
        }
    }
    __syncthreads();

    // ---------------- GEMM3: (16x32) @ (32x2) + bC -------------------------
    if (tid < 32) {
        const int m = tid >> 1;
        const int j = tid & 1;
        float acc = bC[j];
        const float* h2r = &s_h2[m * H2_STRIDE];
        #pragma unroll
        for (int k = 0; k < H2; ++k) acc += h2r[k] * wC[k * 2 + j];
        out[(size_t)(b0 + m) * 2 + j] = acc;
    }
}

extern "C" void kernel_launch(void* const* d_in, const int* in_sizes, int n_in,
                              void* d_out, int out_size, void* d_ws, size_t ws_size,
                              hipStream_t stream) {
    (void)in_sizes; (void)n_in; (void)d_ws; (void)ws_size; (void)out_size;
    const float* x      = (const float*)d_in[0];
    const float* w_step = (const float*)d_in[1];
    const float* b_step = (const float*)d_in[2];
    const float* wA     = (const float*)d_in[3];
    const float* bA     = (const float*)d_in[4];
    const float* wB     = (const float*)d_in[5];
    const float* bB     = (const float*)d_in[6];
    const float* wC     = (const float*)d_in[7];
    const float* bC     = (const float*)d_in[8];
    float* out = (float*)d_out;

    const int blocks = 16384 / ROWS_PER_BLOCK;  // 1024
    fused_struct3_kernel<<<blocks, 256, 0, stream>>>(
        x, w_step, b_step, wA, bA, wB, bB, wC, bC, out);
}